// YOLOv2Loss_53077205844622
// MI455X (gfx1250) — compile-verified
//
#include <hip/hip_runtime.h>

#define NB     128
#define NA     5
#define NC     20
#define NG     52
#define NGT    32
#define CELLS  (NG*NG)        /* 2704  */
#define PRIORS (NA*CELLS)     /* 13520 */
#define NCH    (5+NC)         /* 25    */
#define THRF   0.6f
#define EPSF   1e-5f

typedef float v2f __attribute__((ext_vector_type(2)));
typedef float v8f __attribute__((ext_vector_type(8)));

// IoU between two xyxy boxes with precomputed areas (matches _pair_iou: no EPS).
__device__ __forceinline__ float iou_xyxy(float ax1, float ay1, float ax2, float ay2, float aarea,
                                          float bx1, float by1, float bx2, float by2, float barea)
{
    float ltx = fmaxf(ax1, bx1), lty = fmaxf(ay1, by1);
    float rbx = fminf(ax2, bx2), rby = fminf(ay2, by2);
    float wx = fmaxf(rbx - ltx, 0.0f), wy = fmaxf(rby - lty, 0.0f);
    float inter = wx * wy;
    return inter / (aarea + barea - inter);
}

// Default-box (prior) n -> xyxy + area, exactly as the reference builds it.
__device__ __forceinline__ void db_box(int n, const float* __restrict__ anch,
                                       float& x1, float& y1, float& x2, float& y2, float& area)
{
    int a   = n / CELLS;
    int rem = n - a * CELLS;
    int cy  = rem / NG;
    int cx  = rem - cy * NG;
    float cxc = ((float)cx + 0.5f) / (float)NG;
    float cyc = ((float)cy + 0.5f) / (float)NG;
    float aw = anch[2 * a], ah = anch[2 * a + 1];
    x1 = cxc - aw / 2.0f;  y1 = cyc - ah / 2.0f;
    x2 = cxc + aw / 2.0f;  y2 = cyc + ah / 2.0f;
    area = (x2 - x1) * (y2 - y1);
}

__global__ void init_acc(double* acc) { acc[0] = 0.0; }

// Stage A: per (b, prior): best IoU / first-argmax over the 32 GTs; thresholded assignment.
__global__ void match_per_prior(const float* __restrict__ gtb, const float* __restrict__ anch,
                                int* __restrict__ assign)
{
    int b = blockIdx.y;
    int n = blockIdx.x * 256 + threadIdx.x;
    __shared__ float sgt[NGT * 4];
    if (threadIdx.x < NGT * 4) sgt[threadIdx.x] = gtb[(size_t)b * NGT * 4 + threadIdx.x];
    __syncthreads();
    if (n >= PRIORS) return;

    float dx1, dy1, dx2, dy2, darea;
    db_box(n, anch, dx1, dy1, dx2, dy2, darea);

    float best = -1.0f; int bi = 0;
    #pragma unroll 4
    for (int g = 0; g < NGT; ++g) {
        float gxc = sgt[4*g+0], gyc = sgt[4*g+1], gw = sgt[4*g+2], gh = sgt[4*g+3];
        float gx1 = gxc - gw / 2.0f, gy1 = gyc - gh / 2.0f;
        float gx2 = gxc + gw / 2.0f, gy2 = gyc + gh / 2.0f;
        float garea = (gx2 - gx1) * (gy2 - gy1);
        float v = iou_xyxy(gx1, gy1, gx2, gy2, garea, dx1, dy1, dx2, dy2, darea);
        if (v > best) { best = v; bi = g; }   // strict > : first occurrence (jnp.argmax)
    }
    assign[(size_t)b * PRIORS + n] = (best > THRF) ? bi : -1;
}

// Stage B: per (b, g): argmax over 13520 priors (first-index tiebreak).
__global__ void match_per_gt(const float* __restrict__ gtb, const float* __restrict__ anch,
                             int* __restrict__ best_prior)
{
    int g = blockIdx.x, b = blockIdx.y, tid = threadIdx.x;
    const float* gp = gtb + ((size_t)b * NGT + g) * 4;
    float gxc = gp[0], gyc = gp[1], gw = gp[2], gh = gp[3];
    float gx1 = gxc - gw / 2.0f, gy1 = gyc - gh / 2.0f;
    float gx2 = gxc + gw / 2.0f, gy2 = gyc + gh / 2.0f;
    float garea = (gx2 - gx1) * (gy2 - gy1);

    float best = -1.0f; int bi = PRIORS;
    for (int n = tid; n < PRIORS; n += 256) {
        float dx1, dy1, dx2, dy2, darea;
        db_box(n, anch, dx1, dy1, dx2, dy2, darea);
        float v = iou_xyxy(gx1, gy1, gx2, gy2, garea, dx1, dy1, dx2, dy2, darea);
        if (v > best) { best = v; bi = n; }   // ascending n: keeps first max
    }
    __shared__ float sv[256];
    __shared__ int   si[256];
    sv[tid] = best; si[tid] = bi;
    __syncthreads();
    for (int s = 128; s > 0; s >>= 1) {
        if (tid < s) {
            float v2 = sv[tid + s]; int i2 = si[tid + s];
            if (v2 > sv[tid] || (v2 == sv[tid] && i2 < si[tid])) { sv[tid] = v2; si[tid] = i2; }
        }
        __syncthreads();
    }
    if (tid == 0) best_prior[b * NGT + g] = si[0];
}

// Stage C: forced assignment, sequential g ascending => last GT wins on duplicates
// (matches XLA scatter update order for t.at[best_prior].set(gt)).
__global__ void force_assign(const int* __restrict__ best_prior, int* __restrict__ assign)
{
    int b = threadIdx.x;
    if (b < NB)
        for (int g = 0; g < NGT; ++g)
            assign[(size_t)b * PRIORS + best_prior[b * NGT + g]] = g;
}

// Stage D: main per-cell loss + block reduction (wave-level reduction via V_WMMA_F32_16X16X4_F32).
__global__ void __launch_bounds__(256) loss_main(const float* __restrict__ pred,
                                                 const float* __restrict__ anch,
                                                 const float* __restrict__ gtb,
                                                 const int*   __restrict__ gtl,
                                                 const int*   __restrict__ seen,
                                                 const int*   __restrict__ assign,
                                                 double* __restrict__ acc)
{
    int idx  = blockIdx.x * 256 + threadIdx.x;   // grid exactly tiles NB*PRIORS: no OOB path
    int cell = idx % CELLS;
    int t    = idx / CELLS;
    int a    = t % NA;
    int b    = t / NA;

    const float* base = pred + ((size_t)b * NA * NCH + (size_t)a * NCH) * CELLS + cell;
    float aw = anch[2 * a], ah = anch[2 * a + 1];

    float x    = 1.0f / (1.0f + expf(-base[0 * CELLS]));
    float y    = 1.0f / (1.0f + expf(-base[1 * CELLS]));
    float w    = expf(base[2 * CELLS]) * aw;
    float h    = expf(base[3 * CELLS]) * ah;
    float conf = 1.0f / (1.0f + expf(-base[4 * CELLS]));

    int g = assign[(size_t)b * PRIORS + a * CELLS + cell];
    float pos = (g >= 0) ? 1.0f : 0.0f;
    float tbx = 0.f, tby = 0.f, tbw = 0.f, tbh = 0.f;
    int lab = 0;
    if (g >= 0) {
        const float* gp = gtb + ((size_t)b * NGT + g) * 4;
        tbx = gp[0]; tby = gp[1]; tbw = gp[2]; tbh = gp[3];
        lab = gtl[b * NGT + g];
    }

    // elementwise IoU (with EPS), exactly as _elem_iou
    float p1x = x - w / 2.0f, p1y = y - h / 2.0f, p2x = x + w / 2.0f, p2y = y + h / 2.0f;
    float t1x = tbx - tbw / 2.0f, t1y = tby - tbh / 2.0f, t2x = tbx + tbw / 2.0f, t2y = tby + tbh / 2.0f;
    float ltx = fmaxf(p1x, t1x), lty = fmaxf(p1y, t1y);
    float rbx = fminf(p2x, t2x), rby = fminf(p2y, t2y);
    float wx = fmaxf(rbx - ltx, 0.0f), wy = fmaxf(rby - lty, 0.0f);
    float inter = wx * wy;
    float iou = inter / (w * h + tbw * tbh - inter + EPSF);

    float loss = 0.0f;

    // noobj: (mask * conf)^2
    float mask = (iou <= THRF) ? 1.0f : 0.0f;
    float nv = mask * conf;
    loss += nv * nv;

    // obj: L_OBJ * (pos*conf - iou*pos)^2
    float ov = pos * conf - iou * pos;
    loss += 5.0f * (ov * ov);

    // box: (pos*pred - target)^2
    float d0 = pos * x - tbx, d1 = pos * y - tby, d2 = pos * w - tbw, d3 = pos * h - tbh;
    loss += d0 * d0 + d1 * d1 + d2 * d2 + d3 * d3;

    // prior: L_PRIOR * (neg*pred - neg*anchors_truth)^2   (seen < 12800)
    if (seen[0] < 12800) {
        float nf = 1.0f - pos;
        float half = 0.5f / (float)NG;
        float q0 = nf * x - nf * half;
        float q1 = nf * y - nf * half;
        float q2 = nf * w - nf * aw;
        float q3 = nf * h - nf * ah;
        loss += 0.01f * (q0 * q0 + q1 * q1 + q2 * q2 + q3 * q3);
    }

    // class loss: softmax(pred*pos) vs onehot(label or 0).
    // Negative cells (the vast majority): logits are all zero -> skip the 20 strided
    // channel loads entirely and compute softmax(0) = 1/20 the same way jax does.
    if (g >= 0) {
        float lg[NC];
        #pragma unroll
        for (int c = 0; c < NC; ++c) lg[c] = base[(5 + c) * CELLS];
        float m = lg[0];
        #pragma unroll
        for (int c = 1; c < NC; ++c) m = fmaxf(m, lg[c]);
        float e[NC]; float S = 0.0f;
        #pragma unroll
        for (int c = 0; c < NC; ++c) { e[c] = expf(lg[c] - m); S += e[c]; }
        float cl = 0.0f;
        #pragma unroll
        for (int c = 0; c < NC; ++c) {
            float sm = e[c] / S;
            float tt = (c == lab) ? 1.0f : 0.0f;
            float dd = sm - tt;
            cl += dd * dd;
        }
        loss += cl;
    } else {
        float u = 1.0f / 20.0f;            // softmax of zeros: exp(0)/20
        float cl = 0.0f;
        #pragma unroll
        for (int c = 0; c < NC; ++c) {
            float tt = (c == 0) ? 1.0f : 0.0f;   // onehot(label=0)
            float dd = u - tt;
            cl += dd * dd;
        }
        loss += cl;
    }

    // ---- Block reduction: wave-level sum via V_WMMA_F32_16X16X4_F32 ----
    // A = all-ones (16x4), B vgpr0 = per-lane partial, vgpr1 = 0.
    // D[m,n] = sum_k B[k,n]  (row-independent since A is ones), so the 16 column
    // sums (lanes 0..15, D vgpr0) total exactly the 32 partials + 32 zeros,
    // regardless of the exact slot->element mapping.
    v2f av; av.x = 1.0f; av.y = 1.0f;
    v2f bv; bv.x = loss; bv.y = 0.0f;
    v8f cv = {};
    v8f dv = __builtin_amdgcn_wmma_f32_16x16x4_f32(false, av, false, bv,
                                                   (short)0, cv, false, false);
    __shared__ float wsum[8 * 16];
    int lane = threadIdx.x & 31;
    int wid  = threadIdx.x >> 5;
    if (lane < 16) wsum[wid * 16 + lane] = dv[0];
    __syncthreads();
    if (threadIdx.x == 0) {
        float s = 0.0f;
        for (int i = 0; i < 8 * 16; ++i) s += wsum[i];
        atomicAdd(acc, (double)s);
    }
}

__global__ void finalize(const double* __restrict__ acc, float* __restrict__ out)
{
    out[0] = (float)acc[0];
}

extern "C" void kernel_launch(void* const* d_in, const int* in_sizes, int n_in,
                              void* d_out, int out_size, void* d_ws, size_t ws_size,
                              hipStream_t stream)
{
    const float* pred = (const float*)d_in[0];
    const float* anch = (const float*)d_in[1];
    const float* gtb  = (const float*)d_in[2];
    const int*   gtl  = (const int*)d_in[3];
    const int*   seen = (const int*)d_in[4];

    int* assign     = (int*)d_ws;                         // NB*PRIORS ints (6.92 MB)
    int* best_prior = assign + (size_t)NB * PRIORS;       // NB*NGT ints
    double* acc     = (double*)(best_prior + NB * NGT);   // 8-byte aligned by construction

    init_acc<<<1, 1, 0, stream>>>(acc);
    match_per_prior<<<dim3((PRIORS + 255) / 256, NB), 256, 0, stream>>>(gtb, anch, assign);
    match_per_gt<<<dim3(NGT, NB), 256, 0, stream>>>(gtb, anch, best_prior);
    force_assign<<<1, 128, 0, stream>>>(best_prior, assign);
    loss_main<<<(NB * PRIORS) / 256, 256, 0, stream>>>(pred, anch, gtb, gtl, seen, assign, acc);
    finalize<<<1, 1, 0, stream>>>(acc, (float*)d_out);
}